// EdgeNetwork_81647328297538
// MI455X (gfx1250) — compile-verified
//
#include <hip/hip_runtime.h>

typedef __attribute__((ext_vector_type(2))) float v2f;
typedef __attribute__((ext_vector_type(8))) float v8f;

#define LN_EPS 1e-5f

__device__ __forceinline__ float htanh(float x) {
#if __has_builtin(__builtin_amdgcn_tanhf)
    return __builtin_amdgcn_tanhf(x);        // v_tanh_f32
#elif __has_builtin(__builtin_amdgcn_tanh_f32)
    return __builtin_amdgcn_tanh_f32(x);     // v_tanh_f32
#else
    float r;
    asm volatile("v_tanh_f32 %0, %1" : "=v"(r) : "v"(x));
    return r;
#endif
}

// LayerNorm over 8 lane-local features + affine + tanh
__device__ __forceinline__ void ln_tanh8(const v8f acc, const float* G, const float* Be,
                                         float* h) {
    float mu = 0.0f;
#pragma unroll
    for (int r = 0; r < 8; ++r) mu += acc[r];
    mu *= 0.125f;
    float var = 0.0f;
#pragma unroll
    for (int r = 0; r < 8; ++r) { float d = acc[r] - mu; var += d * d; }
    var *= 0.125f;
    float rs = __builtin_amdgcn_rsqf(var + LN_EPS);   // v_rsq_f32
#pragma unroll
    for (int r = 0; r < 8; ++r) h[r] = htanh((acc[r] - mu) * rs * G[r] + Be[r]);
}

#define WMMA_F32X4(AV, BV, CV) \
    __builtin_amdgcn_wmma_f32_16x16x4_f32(false, (AV), false, (BV), (short)0, (CV), false, false)

__global__ __launch_bounds__(256) void edge_mlp_wmma(
    const float* __restrict__ x,          // [N,16]
    const int*   __restrict__ edge_index, // [2,E]
    const float* __restrict__ vp,         // [G,16]
    const int*   __restrict__ batch,      // [N]
    const float* __restrict__ W1, const float* __restrict__ b1,
    const float* __restrict__ g1, const float* __restrict__ be1,
    const float* __restrict__ W2, const float* __restrict__ b2,
    const float* __restrict__ g2, const float* __restrict__ be2,
    const float* __restrict__ W3, const float* __restrict__ b3,
    const float* __restrict__ g3, const float* __restrict__ be3,
    const float* __restrict__ W4, const float* __restrict__ b4,
    float* __restrict__ out, int E)
{
    const int lane = threadIdx.x & 31;
    const int j    = lane & 15;   // A: output-feature row j ; B/D: edge column m
    const int kh   = lane >> 4;   // K-half selector within a 4-wide K chunk

    // ---- A-matrix fragments: A = W^T, rows j>=8 zero-padded ----
    // chunk c: VGPR0 = A[j][4c+2kh] = W[(4c+2kh)*8 + j], VGPR1 = +1
    float wa1[24];
#pragma unroll
    for (int c = 0; c < 12; ++c) {
        int k0 = 4 * c + 2 * kh;
        wa1[2 * c]     = (j < 8) ? W1[k0 * 8 + j]       : 0.0f;
        wa1[2 * c + 1] = (j < 8) ? W1[(k0 + 1) * 8 + j] : 0.0f;
    }
    float wa2[4], wa3[4];
#pragma unroll
    for (int c = 0; c < 2; ++c) {
        int k0 = 4 * c + 2 * kh;
        wa2[2 * c]     = (j < 8) ? W2[k0 * 8 + j]       : 0.0f;
        wa2[2 * c + 1] = (j < 8) ? W2[(k0 + 1) * 8 + j] : 0.0f;
        wa3[2 * c]     = (j < 8) ? W3[k0 * 8 + j]       : 0.0f;
        wa3[2 * c + 1] = (j < 8) ? W3[(k0 + 1) * 8 + j] : 0.0f;
    }

    // ---- uniform per-feature params (expect SGPRs) ----
    float G1[8], E1[8], G2[8], E2[8], G3[8], E3[8];
    float C1[8], C2[8], C3[8], w4[8];
#pragma unroll
    for (int r = 0; r < 8; ++r) {
        G1[r] = g1[r]; E1[r] = be1[r];
        G2[r] = g2[r]; E2[r] = be2[r];
        G3[r] = g3[r]; E3[r] = be3[r];
        C1[r] = b1[r]; C2[r] = b2[r]; C3[r] = b3[r];
        w4[r] = W4[r];
    }
    const float bias4 = b4[0];

    const int nTiles = (E + 15) >> 4;
    const int stride = (gridDim.x * blockDim.x) >> 5;
    int tile = (blockIdx.x * blockDim.x + threadIdx.x) >> 5;

    for (; tile < nTiles; tile += stride) {
        const int e  = tile * 16 + j;
        const int ec = (e < E) ? e : (E - 1);   // clamp: keep EXEC all-1s for WMMA

        // unsigned 32-bit offsets -> saddr + zext(voffset) global-load form
        const unsigned s  = (unsigned)edge_index[ec];
        const unsigned d  = (unsigned)edge_index[E + ec];
        const unsigned bg = (unsigned)batch[s];
        const unsigned fo = (unsigned)(2 * kh);
        const float* xs = x  + s  * 16u + fo;
        const float* xd = x  + d  * 16u + fo;
        const float* vv = vp + bg * 16u + fo;

        // ---- layer 1: D = W1^T (16x48) x X^T (48x16 edges), bias in C ----
        v8f acc;
#pragma unroll
        for (int r = 0; r < 8; ++r) acc[r] = (lane < 16) ? C1[r] : 0.0f;

#pragma unroll
        for (int c = 0; c < 4; ++c) {            // features 0..15  : x[start]
            v2f bv = *(const v2f*)(xs + 4 * c);
            v2f av = { wa1[2 * c], wa1[2 * c + 1] };
            acc = WMMA_F32X4(av, bv, acc);
        }
#pragma unroll
        for (int c = 0; c < 4; ++c) {            // features 16..31 : x[end]
            v2f bv = *(const v2f*)(xd + 4 * c);
            v2f av = { wa1[2 * (c + 4)], wa1[2 * (c + 4) + 1] };
            acc = WMMA_F32X4(av, bv, acc);
        }
#pragma unroll
        for (int c = 0; c < 4; ++c) {            // features 32..47 : vp[batch[start]]
            v2f bv = *(const v2f*)(vv + 4 * c);
            v2f av = { wa1[2 * (c + 8)], wa1[2 * (c + 8) + 1] };
            acc = WMMA_F32X4(av, bv, acc);
        }

        float h[8];
        ln_tanh8(acc, G1, E1, h);                // lane-local LN + tanh (lane = edge)

        // ---- layer 2: re-marshal h (lane=edge, reg=feature) into B fragments ----
#pragma unroll
        for (int r = 0; r < 8; ++r) acc[r] = (lane < 16) ? C2[r] : 0.0f;
#pragma unroll
        for (int c = 0; c < 2; ++c) {
            float t0 = __shfl(h[4 * c + 2], j, 32);
            float t1 = __shfl(h[4 * c + 3], j, 32);
            v2f bv = { (lane < 16) ? h[4 * c]     : t0,
                       (lane < 16) ? h[4 * c + 1] : t1 };
            v2f av = { wa2[2 * c], wa2[2 * c + 1] };
            acc = WMMA_F32X4(av, bv, acc);
        }
        ln_tanh8(acc, G2, E2, h);

        // ---- layer 3 ----
#pragma unroll
        for (int r = 0; r < 8; ++r) acc[r] = (lane < 16) ? C3[r] : 0.0f;
#pragma unroll
        for (int c = 0; c < 2; ++c) {
            float t0 = __shfl(h[4 * c + 2], j, 32);
            float t1 = __shfl(h[4 * c + 3], j, 32);
            v2f bv = { (lane < 16) ? h[4 * c]     : t0,
                       (lane < 16) ? h[4 * c + 1] : t1 };
            v2f av = { wa3[2 * c], wa3[2 * c + 1] };
            acc = WMMA_F32X4(av, bv, acc);
        }
        ln_tanh8(acc, G3, E3, h);

        // ---- layer 4: lane-local dot with W4 ----
        float o = bias4;
#pragma unroll
        for (int r = 0; r < 8; ++r) o = __builtin_fmaf(h[r], w4[r], o);

        if (lane < 16 && e < E) out[e] = o;
    }
}

extern "C" void kernel_launch(void* const* d_in, const int* in_sizes, int n_in,
                              void* d_out, int out_size, void* d_ws, size_t ws_size,
                              hipStream_t stream) {
    const float* x          = (const float*)d_in[0];
    const int*   edge_index = (const int*)  d_in[1];
    const float* vp         = (const float*)d_in[2];
    const int*   batch      = (const int*)  d_in[3];
    const float* W1 = (const float*)d_in[4];  const float* b1  = (const float*)d_in[5];
    const float* g1 = (const float*)d_in[6];  const float* be1 = (const float*)d_in[7];
    const float* W2 = (const float*)d_in[8];  const float* b2  = (const float*)d_in[9];
    const float* g2 = (const float*)d_in[10]; const float* be2 = (const float*)d_in[11];
    const float* W3 = (const float*)d_in[12]; const float* b3  = (const float*)d_in[13];
    const float* g3 = (const float*)d_in[14]; const float* be3 = (const float*)d_in[15];
    const float* W4 = (const float*)d_in[16]; const float* b4  = (const float*)d_in[17];

    const int E = in_sizes[1] / 2;
    const int nTiles = (E + 15) / 16;
    const int wavesPerBlock = 256 / 32;
    const int tilesPerWave  = 8;   // amortize per-wave weight-marshalling setup
    int blocks = (nTiles + wavesPerBlock * tilesPerWave - 1) / (wavesPerBlock * tilesPerWave);
    if (blocks < 1) blocks = 1;

    edge_mlp_wmma<<<blocks, 256, 0, stream>>>(
        x, edge_index, vp, batch,
        W1, b1, g1, be1, W2, b2, g2, be2, W3, b3, g3, be3, W4, b4,
        (float*)d_out, E);
}